// FPMC_28750511079473
// MI455X (gfx1250) — compile-verified
//
#include <hip/hip_runtime.h>

typedef __attribute__((ext_vector_type(2))) float v2f;
typedef __attribute__((ext_vector_type(8))) float v8f;
typedef __attribute__((ext_vector_type(4))) unsigned v4u;
typedef __attribute__((ext_vector_type(8))) int v8i;
typedef __attribute__((ext_vector_type(4))) int v4i;

#define B_ 1024
#define S_ 200
#define A_ 50000
#define K_ 64
#define UK 256                      // padded K-dim of U rows (194 -> 256)
#define U_FLOATS (B_ * UK)          // ws floats used by U
#define WF_PAD 260                  // LDS row stride for Wf (bank de-conflict)
#define U_PAD 258                   // LDS row stride for U block
// ws float layout: [0, U_FLOATS): U[1024][256]
//                  [U_FLOATS +  0 ..  63]: ILmean
//                  [U_FLOATS + 64 .. 127]: Wdamean
//                  [U_FLOATS + 128]      : bdamean

#if __has_builtin(__builtin_amdgcn_tensor_load_to_lds) && \
    __has_builtin(__builtin_amdgcn_s_wait_tensorcnt)
#define USE_TDM 1
#else
#define USE_TDM 0
#endif

__device__ __forceinline__ v8f wmma_k4(v2f a, v2f b, v8f c) {
  // D = A(16x4 f32) * B(4x16 f32) + C(16x16 f32)
  return __builtin_amdgcn_wmma_f32_16x16x4_f32(false, a, false, b, (short)0, c,
                                               false, false);
}

#if USE_TDM
// Stage a (rows x 256 f32) tile from global to LDS with TDM, adding
// (pad_field+1) DWORDs of padding per 256-DWORD row (pad_interval code 7).
// lds_off = LDS byte address (low 32 bits of generic shared pointer).
__device__ __forceinline__ void tdm_stage_rows256(unsigned lds_off,
                                                  const void* gptr,
                                                  unsigned rows,
                                                  unsigned pad_field) {
  const unsigned long long ga = (unsigned long long)gptr;
  v4u g0;
  g0[0] = 1u;                                   // count=1, user descriptor
  g0[1] = lds_off;                              // lds_addr
  g0[2] = (unsigned)(ga & 0xffffffffu);         // global_addr[31:0]
  g0[3] = (unsigned)((ga >> 32) & 0x01ffffffu)  // global_addr[56:32]
          | (2u << 30);                         // type=2 ("image")
  v8i g1;
  // data_size=2 (4B), pad_enable, pad_interval=7 (256 DWORDs), pad_amount
  g1[0] = (int)((2u << 16) | (1u << 20) | (7u << 22) | (pad_field << 25));
  g1[1] = (int)((256u & 0xffffu) << 16);        // tensor_dim0=256 (bits 79:48)
  g1[2] = (int)((rows & 0xffffu) << 16);        // dim0 hi=0, tensor_dim1 lo
  g1[3] = (int)((rows >> 16) | (256u << 16));   // tensor_dim1 hi, tile_dim0=256
  g1[4] = (int)(rows & 0xffffu);                // tile_dim1=rows, tile_dim2=0
  g1[5] = 256;                                  // tensor_dim0_stride lo
  g1[6] = 0;                                    // stride hi, dim1_stride lo
  g1[7] = 0;
  v4i gz = {0, 0, 0, 0};
#if defined(__clang_major__) && (__clang_major__ >= 23)
  v8i gz8 = {0, 0, 0, 0, 0, 0, 0, 0};
  __builtin_amdgcn_tensor_load_to_lds(g0, g1, gz, gz, gz8, 0);
#else
  __builtin_amdgcn_tensor_load_to_lds(g0, g1, gz, gz, 0);
#endif
}
#endif

// ---------------------------------------------------------------------------
// Kernel 0: column means of IL[1:], Wda, and mean of bda.
// ---------------------------------------------------------------------------
__global__ void fpmc_means(const float* __restrict__ IL,
                           const float* __restrict__ Wda,
                           const float* __restrict__ bda,
                           float* __restrict__ ws) {
  __shared__ float red[256];
  const int bid = blockIdx.x;
  const int tid = threadIdx.x;
  float sum = 0.f;
  if (bid < 64) {
    const int j = bid;
    for (int a = 1 + tid; a <= A_; a += 256) sum += IL[(long)a * 64 + j];
  } else if (bid < 128) {
    const int j = bid - 64;
    for (int a = tid; a < A_; a += 256) sum += Wda[(long)a * 64 + j];
  } else {
    for (int a = tid; a < A_; a += 256) sum += bda[a];
  }
  red[tid] = sum;
  __syncthreads();
  for (int s = 128; s > 0; s >>= 1) {
    if (tid < s) red[tid] += red[tid + s];
    __syncthreads();
  }
  if (tid == 0) {
    float* dst = ws + U_FLOATS;
    dst[bid] = red[0] / (float)A_;
  }
}

// ---------------------------------------------------------------------------
// Kernel 1: one workgroup per batch b. Wf staged to LDS (TDM if available),
// fused = feats @ Wf^T + bf via WMMA (never stored), weighted-sum -> U row.
// ---------------------------------------------------------------------------
__global__ void fpmc_batch(const int* __restrict__ days,
                           const int* __restrict__ times,
                           const int* __restrict__ devices,
                           const int* __restrict__ actions,
                           const float* __restrict__ masks,
                           const float* __restrict__ day_tab,
                           const float* __restrict__ time_tab,
                           const float* __restrict__ dev_tab,
                           const float* __restrict__ act_tab,
                           const float* __restrict__ LI,
                           const float* __restrict__ LD,
                           const float* __restrict__ Wf,
                           const float* __restrict__ bf,
                           float* __restrict__ ws) {
  const int b = blockIdx.x;
  const int tid = threadIdx.x;   // 0..127
  const int lane = tid & 31;
  const int wave = tid >> 5;     // 0..3
  const int m = lane & 15;       // row (A) / col (B) within 16-tile
  const int half = lane >> 4;    // K-pair selector

  __shared__ float WfL[64 * WF_PAD];  // Wf[64][256] staged, padded rows
  __shared__ float accum[64];         // sum_s w[s]*fused[s,:]
  __shared__ float wts[S_];           // per-position weights
  __shared__ int sh_L;
  __shared__ float sh_denom;
  __shared__ int sh_la, sh_ld;

#if USE_TDM
  if (tid == 0) {
    tdm_stage_rows256((unsigned)(unsigned long long)(void*)WfL, Wf, 64u,
                      3u /* +4 DWORDs pad -> stride 260 */);
  }
#else
  for (int idx = tid; idx < 64 * 64; idx += 128) {
    const int r = idx >> 6;
    const int c4 = (idx & 63) << 2;
    *(float4*)(WfL + r * WF_PAD + c4) =
        *(const float4*)(Wf + (long)r * 256 + c4);
  }
#endif

  if (tid < 64) accum[tid] = 0.f;
  if (tid == 0) {
    int L = 0;
    for (int s = 0; s < S_; ++s) L += (masks[(long)b * S_ + s] > 0.5f) ? 1 : 0;
    sh_L = L;
  }
  __syncthreads();
  const int L = sh_L;
  for (int s = tid; s < S_; s += 128)
    wts[s] = (s < L) ? __expf((float)(s + 1 - L)) : 0.f;
  __syncthreads();
  if (tid == 0) {
    float d = 0.f;
    for (int s = 0; s < S_; ++s) d += wts[s];
    sh_denom = fmaxf(d, 1e-9f);
  }
  __syncthreads();
  const float inv_denom = 1.0f / sh_denom;
  for (int s = tid; s < S_; s += 128) wts[s] *= inv_denom;
#if USE_TDM
  __builtin_amdgcn_s_wait_tensorcnt(0);  // no-op for waves that didn't issue
#endif
  __syncthreads();

  // LDS row pointers for the 4 output col-tiles (B fragment: Wf[n0+m, k..k+1])
  const float* wfl0 = WfL + (0 + m) * WF_PAD + 2 * half;
  const float* wfl1 = WfL + (16 + m) * WF_PAD + 2 * half;
  const float* wfl2 = WfL + (32 + m) * WF_PAD + 2 * half;
  const float* wfl3 = WfL + (48 + m) * WF_PAD + 2 * half;

  const int ntiles = (S_ + 15) / 16;  // 13
  for (int t = wave; t < ntiles; t += 4) {
    const int s0 = t * 16;
    int s = s0 + m;
    int sidx = (s < S_) ? s : (S_ - 1);
    const long off = (long)b * S_ + sidx;
    const float* basep[4];
    basep[0] = day_tab + (long)days[off] * 64 + 2 * half;
    basep[1] = time_tab + (long)times[off] * 64 + 2 * half;
    basep[2] = dev_tab + (long)devices[off] * 64 + 2 * half;
    basep[3] = act_tab + (long)actions[off] * 64 + 2 * half;

    v8f c0 = {}, c1 = {}, c2 = {}, c3 = {};
#pragma unroll
    for (int seg = 0; seg < 4; ++seg) {
      const float* ap = basep[seg];
      const int so = seg * 64;
#pragma unroll 2
      for (int k = 0; k < 64; k += 4) {
        v2f af = *(const v2f*)(ap + k);
        v2f b0 = *(const v2f*)(wfl0 + so + k);
        v2f b1 = *(const v2f*)(wfl1 + so + k);
        v2f b2 = *(const v2f*)(wfl2 + so + k);
        v2f b3 = *(const v2f*)(wfl3 + so + k);
        c0 = wmma_k4(af, b0, c0);
        c1 = wmma_k4(af, b1, c1);
        c2 = wmma_k4(af, b2, c2);
        c3 = wmma_k4(af, b3, c3);
      }
    }
    const float bf0 = bf[0 + m], bf1 = bf[16 + m], bf2 = bf[32 + m],
                bf3 = bf[48 + m];
    float p0 = 0.f, p1 = 0.f, p2 = 0.f, p3 = 0.f;
#pragma unroll
    for (int j = 0; j < 8; ++j) {
      const int row = s0 + j + 8 * half;
      const float w = (row < S_) ? wts[row] : 0.f;
      p0 += (c0[j] + bf0) * w;
      p1 += (c1[j] + bf1) * w;
      p2 += (c2[j] + bf2) * w;
      p3 += (c3[j] + bf3) * w;
    }
    atomicAdd(&accum[0 + m], p0);
    atomicAdd(&accum[16 + m], p1);
    atomicAdd(&accum[32 + m], p2);
    atomicAdd(&accum[48 + m], p3);
  }
  __syncthreads();

  if (tid == 0) {
    const int li = (L > 0) ? (L - 1) : 0;
    sh_la = (L > 0) ? actions[(long)b * S_ + li] : 0;
    sh_ld = (L > 0) ? devices[(long)b * S_ + li] : 0;
  }
  __syncthreads();

  const float* ILm = ws + U_FLOATS;
  const float* WDm = ILm + 64;
  float* U = ws + (long)b * UK;
  if (tid < 64) {
    U[tid] = LI[(long)sh_la * 64 + tid];
    U[64 + tid] = LD[(long)sh_ld * 64 + tid];
    U[128 + tid] = accum[tid] / (float)S_;   // seq_avg[0:64]
  }
  if (tid < 62) U[194 + tid] = 0.f;          // pad 194..255
  if (tid == 0) {
    float am = 0.f, dm = 0.f;
    for (int k = 0; k < 64; ++k) {
      am += LI[(long)sh_la * 64 + k] * ILm[k];
      dm += LD[(long)sh_ld * 64 + k] * WDm[k];
    }
    U[192] = am;                 // a_mean
    U[193] = dm + ILm[128];      // d_mean = LD.Wdamean + bdamean
  }
}

// ---------------------------------------------------------------------------
// Kernel 2: out[1024, 50000] = U @ V^T + bias, V streamed from IL/Wda/Wo.
// 256 threads = 8 waves; each wave: 32 b-rows x 16 a-cols (two C tiles share
// every B fragment). Block tile = 32 x 128. U block staged to LDS (TDM).
// ---------------------------------------------------------------------------
__global__ void fpmc_out(const float* __restrict__ IL,
                         const float* __restrict__ Wda,
                         const float* __restrict__ Wo,
                         const float* __restrict__ bo,
                         const float* __restrict__ bda,
                         const float* __restrict__ ws,
                         float* __restrict__ out) {
  const int tid = threadIdx.x;
  const int lane = tid & 31;
  const int wave = tid >> 5;               // 0..7
  const int rblk = blockIdx.x * 32;        // base row of this block
  const int a0 = blockIdx.y * 128 + wave * 16;
  const int m = lane & 15;
  const int half = lane >> 4;

  __shared__ float UL[32 * U_PAD];         // U[rblk..rblk+31][0..255] staged

#if USE_TDM
  if (tid == 0) {
    tdm_stage_rows256((unsigned)(unsigned long long)(void*)UL,
                      ws + (long)rblk * UK, 32u,
                      1u /* +2 DWORDs pad -> stride 258 */);
  }
  __builtin_amdgcn_s_wait_tensorcnt(0);
#else
  for (int idx = tid; idx < 32 * 128; idx += 256) {
    const int r = idx >> 7;
    const int c2 = (idx & 127) * 2;
    *(v2f*)(UL + r * U_PAD + c2) =
        *(const v2f*)(ws + (long)(rblk + r) * UK + c2);
  }
#endif
  __syncthreads();

  const int acol = a0 + m;
  const bool avalid = (acol < A_);
  const int acl = avalid ? acol : (A_ - 1);

  // V row segments for this lane's column
  const float* vb0 = IL + (long)(1 + acl) * 64 + 2 * half;
  const float* vb1 = Wda + (long)acl * 64 + 2 * half;
  const float* vb2 = Wo + (long)acl * 66 + 2 * half;
  __builtin_prefetch(vb0, 0, 0);
  __builtin_prefetch(vb1, 0, 0);
  __builtin_prefetch(vb2, 0, 0);

  const float* ua_lo = UL + (0 + m) * U_PAD + 2 * half;   // rows 0..15
  const float* ua_hi = UL + (16 + m) * U_PAD + 2 * half;  // rows 16..31

  v8f clo = {}, chi = {};
#pragma unroll 4
  for (int k = 0; k < 64; k += 4) {        // segment 0: LI . IL[1+a]
    v2f bfv = *(const v2f*)(vb0 + k);
    v2f alo = *(const v2f*)(ua_lo + k);
    v2f ahi = *(const v2f*)(ua_hi + k);
    clo = wmma_k4(alo, bfv, clo);
    chi = wmma_k4(ahi, bfv, chi);
  }
#pragma unroll 4
  for (int k = 0; k < 64; k += 4) {        // segment 1: LD . Wda[a]
    v2f bfv = *(const v2f*)(vb1 + k);
    v2f alo = *(const v2f*)(ua_lo + 64 + k);
    v2f ahi = *(const v2f*)(ua_hi + 64 + k);
    clo = wmma_k4(alo, bfv, clo);
    chi = wmma_k4(ahi, bfv, chi);
  }
#pragma unroll 4
  for (int k = 0; k < 64; k += 4) {        // segment 2: seq_avg . Wo[a,0:64]
    v2f bfv = *(const v2f*)(vb2 + k);
    v2f alo = *(const v2f*)(ua_lo + 128 + k);
    v2f ahi = *(const v2f*)(ua_hi + 128 + k);
    clo = wmma_k4(alo, bfv, clo);
    chi = wmma_k4(ahi, bfv, chi);
  }

  // epilogue: rank-2 (a_mean, d_mean) terms + biases, store both row tiles
  const float wo64 = Wo[(long)acl * 66 + 64];
  const float wo65 = Wo[(long)acl * 66 + 65];
  const float bias = bda[acl] + bo[acl];
#pragma unroll
  for (int j = 0; j < 8; ++j) {
    const int rlo = j + 8 * half;
    const int rhi = 16 + j + 8 * half;
    const float vlo = clo[j] + UL[rlo * U_PAD + 192] * wo64 +
                      UL[rlo * U_PAD + 193] * wo65 + bias;
    const float vhi = chi[j] + UL[rhi * U_PAD + 192] * wo64 +
                      UL[rhi * U_PAD + 193] * wo65 + bias;
    if (avalid) {
      out[(long)(rblk + rlo) * A_ + acol] = vlo;
      out[(long)(rblk + rhi) * A_ + acol] = vhi;
    }
  }
}

// ---------------------------------------------------------------------------
extern "C" void kernel_launch(void* const* d_in, const int* in_sizes, int n_in,
                              void* d_out, int out_size, void* d_ws,
                              size_t ws_size, hipStream_t stream) {
  const int* days = (const int*)d_in[0];
  const int* times = (const int*)d_in[1];
  const int* devices = (const int*)d_in[2];
  const int* actions = (const int*)d_in[3];
  const float* masks = (const float*)d_in[4];
  const float* day_tab = (const float*)d_in[5];
  const float* time_tab = (const float*)d_in[6];
  const float* dev_tab = (const float*)d_in[7];
  const float* act_tab = (const float*)d_in[8];
  const float* IL = (const float*)d_in[9];
  const float* LI = (const float*)d_in[10];
  const float* LD = (const float*)d_in[11];
  const float* Wf = (const float*)d_in[12];
  const float* bf = (const float*)d_in[13];
  const float* Wo = (const float*)d_in[14];
  const float* bo = (const float*)d_in[15];
  const float* Wda = (const float*)d_in[16];
  const float* bda = (const float*)d_in[17];

  float* out = (float*)d_out;
  float* ws = (float*)d_ws;

  fpmc_means<<<129, 256, 0, stream>>>(IL, Wda, bda, ws);
  fpmc_batch<<<B_, 128, 0, stream>>>(days, times, devices, actions, masks,
                                     day_tab, time_tab, dev_tab, act_tab, LI,
                                     LD, Wf, bf, ws);
  fpmc_out<<<dim3(32, (A_ + 127) / 128), 256, 0, stream>>>(IL, Wda, Wo, bo,
                                                           bda, ws, out);
}